// GraphConnectivityDecoder_13211319402652
// MI455X (gfx1250) — compile-verified
//
#include <hip/hip_runtime.h>
#include <hip/hip_bf16.h>

#define NNODES 19
#define NEG_SLOPE 0.2f

typedef __attribute__((ext_vector_type(2))) float v2f;
typedef __attribute__((ext_vector_type(8))) float v8f;

// ---------------------------------------------------------------------------
// out[19, N] = act( A[19,K] @ W[N,K]^T + bias[N] ),  act: 0=none, 1=relu
// One wave per 16x16 tile, V_WMMA_F32_16X16X4_F32 along K in steps of 4.
// A-frag (16x4):  lanes 0-15: {K=k,k+1} row M=lane; lanes 16-31: {K=k+2,k+3}.
// B-frag (4x16):  lanes 0-15: rows K=k,k+1 col N=lane; lanes 16-31: K=k+2,k+3.
// C/D: vgpr r -> M = r + 8*(lane>=16), N = lane&15.
//
// All loads are unconditional from clamped (always-valid) addresses; bounds
// guards lower to v_cndmask, not exec-masked branches. The K tail (K%4!=0)
// is peeled as one guarded, wave-uniform iteration so the main loop is a
// tight load/load/wmma loop and EXEC is all-1s at every WMMA.
// ---------------------------------------------------------------------------
__global__ __launch_bounds__(32) void wmma_gemm_kernel(
    const float* __restrict__ A, const float* __restrict__ W,
    const float* __restrict__ bias, float* __restrict__ out,
    int K, int N, int act)
{
    const int lane = threadIdx.x & 31;
    const int half = lane >> 4;        // 0 or 1
    const int l16  = lane & 15;
    const int n0 = blockIdx.x * 16;
    const int m0 = blockIdx.y * 16;

    const int  rowA   = m0 + l16;
    const bool rowOk  = rowA < NNODES;
    const int  rowSafe = rowOk ? rowA : (NNODES - 1);
    const float* __restrict__ Arow = A + (size_t)rowSafe * K;
    const float* __restrict__ Wrow = W + (size_t)(n0 + l16) * K;

    const int Kmain = K & ~3;
    v8f c = {};

    for (int k = 0; k < Kmain; k += 4) {
        const int ka = k + 2 * half;
        const float a0 = Arow[ka];
        const float a1 = Arow[ka + 1];
        v2f a, b;
        b.x = Wrow[ka];
        b.y = Wrow[ka + 1];
        a.x = rowOk ? a0 : 0.0f;
        a.y = rowOk ? a1 : 0.0f;
        c = __builtin_amdgcn_wmma_f32_16x16x4_f32(false, a, false, b,
                                                  (short)0, c, false, false);
    }

    if (Kmain < K) {                   // wave-uniform tail, guarded via selects
        const int  ka  = Kmain + 2 * half;
        const bool v0  = ka     < K;
        const bool v1  = ka + 1 < K;
        const int  ka0 = v0 ? ka     : (K - 1);
        const int  ka1 = v1 ? ka + 1 : (K - 1);
        const float a0 = Arow[ka0];
        const float a1 = Arow[ka1];
        const float b0 = Wrow[ka0];
        const float b1 = Wrow[ka1];
        v2f a, b;
        a.x = (rowOk && v0) ? a0 : 0.0f;
        a.y = (rowOk && v1) ? a1 : 0.0f;
        b.x = v0 ? b0 : 0.0f;
        b.y = v1 ? b1 : 0.0f;
        c = __builtin_amdgcn_wmma_f32_16x16x4_f32(false, a, false, b,
                                                  (short)0, c, false, false);
    }

    const float bv = bias[n0 + l16];
#pragma unroll
    for (int r = 0; r < 8; ++r) {
        const int row = m0 + r + 8 * half;
        if (row < NNODES) {
            float v = c[r] + bv;
            if (act == 1) v = v > 0.0f ? v : 0.0f;
            out[(size_t)row * N + n0 + l16] = v;
        }
    }
}

// ---------------------------------------------------------------------------
// GATv2 attention + aggregation. Graph is complete + self-loops, so the
// segment softmax over dst is a dense 19x19 row softmax (deterministic).
//   L[i][j]   = att . leaky_relu(xl[j] + xr[i])
//   alpha     = softmax_j(L[i][:])
//   h[i,:]    = sum_j alpha[i][j] * xl[j,:] + b
//   alpha_out (optional) emitted in reference edge order: 342 edges then loops
// ---------------------------------------------------------------------------
__global__ __launch_bounds__(384) void gat_kernel(
    const float* __restrict__ xl, const float* __restrict__ xr,
    const float* __restrict__ att, const float* __restrict__ bvec,
    const int* __restrict__ esrc, const int* __restrict__ edst, int E,
    float* __restrict__ hout, float* __restrict__ alpha_out, int D)
{
    __shared__ float Lsh[NNODES * NNODES];
    __shared__ float Ash[NNODES * NNODES];
    __shared__ float mxsh[NNODES];
    __shared__ float invsh[NNODES];
    const int t = threadIdx.x;

    for (int p = t; p < NNODES * NNODES; p += blockDim.x) {
        const int i = p / NNODES, j = p % NNODES;   // i = dst, j = src
        const float* __restrict__ xlj = xl + (size_t)j * D;
        const float* __restrict__ xri = xr + (size_t)i * D;
        float acc = 0.0f;
        for (int d = 0; d < D; ++d) {
            const float s  = xlj[d] + xri[d];
            const float lr = s > 0.0f ? s : NEG_SLOPE * s;
            acc += att[d] * lr;
        }
        Lsh[p] = acc;
    }
    __syncthreads();

    if (t < NNODES) {
        float mx = -3.0e38f;
        for (int j = 0; j < NNODES; ++j) mx = fmaxf(mx, Lsh[t * NNODES + j]);
        float s = 0.0f;
        for (int j = 0; j < NNODES; ++j) s += expf(Lsh[t * NNODES + j] - mx);
        mxsh[t]  = mx;
        invsh[t] = 1.0f / s;
    }
    __syncthreads();

    for (int p = t; p < NNODES * NNODES; p += blockDim.x) {
        const int i = p / NNODES;
        Ash[p] = expf(Lsh[p] - mxsh[i]) * invsh[i];
    }
    __syncthreads();

    for (int idx = t; idx < NNODES * D; idx += blockDim.x) {
        const int i = idx / D, d = idx % D;
        float acc = bvec[d];
        for (int j = 0; j < NNODES; ++j)
            acc += Ash[i * NNODES + j] * xl[(size_t)j * D + d];
        hout[idx] = acc;
    }

    if (alpha_out != nullptr) {
        const int Etot = E + NNODES;
        for (int e = t; e < Etot; e += blockDim.x) {
            int s, dd;
            if (e < E) { s = esrc[e]; dd = edst[e]; }
            else       { s = dd = e - E; }
            alpha_out[e] = Ash[dd * NNODES + s];
        }
    }
}

// ---------------------------------------------------------------------------
// GIN neighborhood aggregation (eps=0): out[i,:] = x[i,:] + sum_{dst[e]==i} x[src[e],:]
// Deterministic per-output scan of the edge list (no float atomics).
// ---------------------------------------------------------------------------
__global__ void gin_agg_kernel(const float* __restrict__ x, float* __restrict__ out,
                               const int* __restrict__ esrc, const int* __restrict__ edst,
                               int E, int C)
{
    const int idx = blockIdx.x * blockDim.x + threadIdx.x;
    if (idx >= NNODES * C) return;
    const int i = idx / C, col = idx % C;
    float acc = x[idx];
    for (int e = 0; e < E; ++e)
        if (edst[e] == i) acc += x[(size_t)esrc[e] * C + col];
    out[idx] = acc;
}

// ---------------------------------------------------------------------------
// gf = h2 + mmse*mmse_W + mmse_b ; compressed[i][j] = sigmoid(gf_i . gf_j)
// ---------------------------------------------------------------------------
__global__ __launch_bounds__(384) void decode_kernel(
    const float* __restrict__ h2, const float* __restrict__ mmse,
    const float* __restrict__ mW, const float* __restrict__ mb,
    float* __restrict__ comp_out)
{
    __shared__ float gf[NNODES * 512];
    const float m0 = mmse[0];
    for (int idx = threadIdx.x; idx < NNODES * 512; idx += blockDim.x) {
        const int d = idx % 512;
        gf[idx] = h2[idx] + m0 * mW[d] + mb[d];
    }
    __syncthreads();
    for (int p = threadIdx.x; p < NNODES * NNODES; p += blockDim.x) {
        const int i = p / NNODES, j = p % NNODES;
        float acc = 0.0f;
        for (int d = 0; d < 512; ++d) acc += gf[i * 512 + d] * gf[j * 512 + d];
        comp_out[p] = 1.0f / (1.0f + expf(-acc));
    }
}

// ---------------------------------------------------------------------------
// pred = sigmoid( dot(g.flatten(), pred_W) + pred_b )
// ---------------------------------------------------------------------------
__global__ __launch_bounds__(256) void pred_kernel(
    const float* __restrict__ g, const float* __restrict__ pW,
    const float* __restrict__ pb, float* __restrict__ outp)
{
    __shared__ float red[256];
    float acc = 0.0f;
    for (int i = threadIdx.x; i < NNODES * 512; i += 256) acc += g[i] * pW[i];
    red[threadIdx.x] = acc;
    __syncthreads();
    for (int s = 128; s > 0; s >>= 1) {
        if (threadIdx.x < s) red[threadIdx.x] += red[threadIdx.x + s];
        __syncthreads();
    }
    if (threadIdx.x == 0) outp[0] = 1.0f / (1.0f + expf(-(red[0] + pb[0])));
}

// ---------------------------------------------------------------------------
extern "C" void kernel_launch(void* const* d_in, const int* in_sizes, int n_in,
                              void* d_out, int out_size, void* d_ws, size_t ws_size,
                              hipStream_t stream)
{
    const float* x       = (const float*)d_in[0];   // [19,1025]
    const float* mmse    = (const float*)d_in[1];   // [1]
    const int*   eidx    = (const int*)d_in[2];     // [2,E]
    const float* g1_Wl   = (const float*)d_in[3];
    const float* g1_bl   = (const float*)d_in[4];
    const float* g1_Wr   = (const float*)d_in[5];
    const float* g1_br   = (const float*)d_in[6];
    const float* g1_att  = (const float*)d_in[7];
    const float* g1_b    = (const float*)d_in[8];
    const float* g2_Wl   = (const float*)d_in[9];
    const float* g2_bl   = (const float*)d_in[10];
    const float* g2_Wr   = (const float*)d_in[11];
    const float* g2_br   = (const float*)d_in[12];
    const float* g2_att  = (const float*)d_in[13];
    const float* g2_b    = (const float*)d_in[14];
    const float* mmse_W  = (const float*)d_in[15];
    const float* mmse_b  = (const float*)d_in[16];
    const float* gin1_W1 = (const float*)d_in[17];
    const float* gin1_b1 = (const float*)d_in[18];
    const float* gin1_W2 = (const float*)d_in[19];
    const float* gin1_b2 = (const float*)d_in[20];
    const float* gin2_W1 = (const float*)d_in[21];
    const float* gin2_b1 = (const float*)d_in[22];
    const float* gin2_W2 = (const float*)d_in[23];
    const float* gin2_b2 = (const float*)d_in[24];
    const float* pred_W  = (const float*)d_in[25];
    const float* pred_b  = (const float*)d_in[26];

    const int E = in_sizes[2] / 2;                  // 342
    const int* esrc = eidx;
    const int* edst = eidx + E;

    float* out = (float*)d_out;                     // [361 comp][361 alpha][1 pred]
    float* comp_out  = out;
    float* alpha_out = out + NNODES * NNODES;
    float* pred_out  = out + 2 * NNODES * NNODES;

    // workspace layout (floats)
    float* ws  = (float*)d_ws;
    float* hx  = ws;                  // [19,1025]
    float* xl1 = hx  + 19 * 1025;     // [19,512]
    float* xr1 = xl1 + 19 * 512;
    float* h1  = xr1 + 19 * 512;
    float* xl2 = h1  + 19 * 512;
    float* xr2 = xl2 + 19 * 512;
    float* h2  = xr2 + 19 * 512;
    float* t1  = h2  + 19 * 512;      // [19,256]
    float* gb1 = t1  + 19 * 256;      // [19,256]
    float* hg  = gb1 + 19 * 256;      // [19,256]
    float* t2  = hg  + 19 * 256;      // [19,512]
    float* gb2 = t2  + 19 * 512;      // [19,512]
    (void)ws_size; (void)n_in; (void)out_size;

    const dim3 blk32(32);
    const dim3 grid512(512 / 16, 2);  // N=512, 2 row tiles
    const dim3 grid256(256 / 16, 2);  // N=256

    // ---- GATv2 layer 1 (K=1025) ----
    wmma_gemm_kernel<<<grid512, blk32, 0, stream>>>(x, g1_Wl, g1_bl, xl1, 1025, 512, 0);
    wmma_gemm_kernel<<<grid512, blk32, 0, stream>>>(x, g1_Wr, g1_br, xr1, 1025, 512, 0);
    gat_kernel<<<1, 384, 0, stream>>>(xl1, xr1, g1_att, g1_b, esrc, edst, E, h1, alpha_out, 512);

    // ---- GATv2 layer 2 (K=512) ----
    wmma_gemm_kernel<<<grid512, blk32, 0, stream>>>(h1, g2_Wl, g2_bl, xl2, 512, 512, 0);
    wmma_gemm_kernel<<<grid512, blk32, 0, stream>>>(h1, g2_Wr, g2_br, xr2, 512, 512, 0);
    gat_kernel<<<1, 384, 0, stream>>>(xl2, xr2, g2_att, g2_b, esrc, edst, E, h2, nullptr, 512);

    // ---- mmse conditioning + connectivity decode ----
    decode_kernel<<<1, 384, 0, stream>>>(h2, mmse, mmse_W, mmse_b, comp_out);

    // ---- GIN classifier ----
    gin_agg_kernel<<<(19 * 1025 + 255) / 256, 256, 0, stream>>>(x, hx, esrc, edst, E, 1025);
    wmma_gemm_kernel<<<grid256, blk32, 0, stream>>>(hx, gin1_W1, gin1_b1, t1, 1025, 256, 1);
    wmma_gemm_kernel<<<grid256, blk32, 0, stream>>>(t1, gin1_W2, gin1_b2, gb1, 256, 256, 1); // outer relu fused
    gin_agg_kernel<<<(19 * 256 + 255) / 256, 256, 0, stream>>>(gb1, hg, esrc, edst, E, 256);
    wmma_gemm_kernel<<<grid512, blk32, 0, stream>>>(hg, gin2_W1, gin2_b1, t2, 256, 512, 1);
    wmma_gemm_kernel<<<grid512, blk32, 0, stream>>>(t2, gin2_W2, gin2_b2, gb2, 512, 512, 0);
    pred_kernel<<<1, 256, 0, stream>>>(gb2, pred_W, pred_b, pred_out);
}